// SelfAttention_75703093559679
// MI455X (gfx1250) — compile-verified
//
#include <hip/hip_runtime.h>
#include <hip/hip_bf16.h>

typedef float v2f __attribute__((ext_vector_type(2)));
typedef float v8f __attribute__((ext_vector_type(8)));
typedef int   v4i __attribute__((ext_vector_type(4)));

typedef __attribute__((address_space(1))) v4i g_v4i;   // global 16B chunk
typedef __attribute__((address_space(3))) v4i l_v4i;   // LDS    16B chunk

#define DIM_ 1536
#define S_   5400
#define SPAD 5408
#define NH_  12
#define HD_  128

__device__ __forceinline__ void async_copy16(const float* gp, float* lp) {
  __builtin_amdgcn_global_load_async_to_lds_b128(
      (g_v4i*)(float*)gp, (l_v4i*)lp, 0, 0);
}
#define WAIT_ASYNC() asm volatile("s_wait_asynccnt 0x0" ::: "memory")

// ---------------------------------------------------------------------------
// GEMM: C[M x N] = X[M x K] @ W[N x K]^T + bias[N]   (fp32, WMMA f32_16x16x4)
// Block: 256 thr (8 waves), tile 128x128; wave tile 64x32 (4x2 16x16 accs).
// Double-buffered LDS fed by GLOBAL_LOAD_ASYNC_TO_LDS_B128 (ASYNCcnt).
// ---------------------------------------------------------------------------
#define GP 20  // LDS pitch (floats): rows 16B-aligned, frag reads conflict-free

__global__ __launch_bounds__(256)
void gemm_xwT_bias(const float* __restrict__ X, const float* __restrict__ W,
                   const float* __restrict__ bias, float* __restrict__ C,
                   int M, int N, int K) {
  __shared__ float Xs[2][128 * GP];
  __shared__ float Ws[2][128 * GP];

  const int tid   = threadIdx.x;
  const int lane  = tid & 31;
  const int wid   = tid >> 5;        // 0..7
  const int waveM = wid >> 2;        // 0..1  -> 64-row slab
  const int waveN = wid & 3;         // 0..3  -> 32-col slab
  const int bm    = blockIdx.x * 128;
  const int bn    = blockIdx.y * 128;
  const int idx   = lane & 15;
  const int half  = (lane >> 4) << 1; // 0 or 2 (K split of the frag)

  v8f vzero = {};
  v8f acc[4][2];
  #pragma unroll
  for (int i = 0; i < 4; ++i)
    #pragma unroll
    for (int j = 0; j < 2; ++j) acc[i][j] = vzero;

  const int lrow = tid >> 2;        // 0..63
  const int lcol = (tid & 3) * 4;   // 0,4,8,12

  auto issue_tile = [&](int kt, int buf) {
    #pragma unroll
    for (int p = 0; p < 2; ++p) {
      int r  = lrow + p * 64;
      int gm = bm + r;
      if (gm < M)
        async_copy16(X + (size_t)gm * K + kt + lcol, &Xs[buf][r * GP + lcol]);
      int gn = bn + r;
      if (gn < N)
        async_copy16(W + (size_t)gn * K + kt + lcol, &Ws[buf][r * GP + lcol]);
    }
  };

  const int T = K / 16;
  issue_tile(0, 0);

  for (int t = 0; t < T; ++t) {
    WAIT_ASYNC();        // own slab-t copies landed in LDS
    __syncthreads();     // everyone's copies landed; prev compute fully drained
    if (t + 1 < T) issue_tile((t + 1) * 16, (t + 1) & 1);  // overlap next slab

    const float* Xb = Xs[t & 1];
    const float* Wb = Ws[t & 1];
    #pragma unroll
    for (int kk = 0; kk < 16; kk += 4) {
      v2f a[4], b[2];
      #pragma unroll
      for (int i = 0; i < 4; ++i) {
        int r = waveM * 64 + i * 16 + idx;
        a[i].x = Xb[r * GP + kk + half];
        a[i].y = Xb[r * GP + kk + half + 1];
      }
      #pragma unroll
      for (int j = 0; j < 2; ++j) {
        int r = waveN * 32 + j * 16 + idx;
        b[j].x = Wb[r * GP + kk + half];
        b[j].y = Wb[r * GP + kk + half + 1];
      }
      #pragma unroll
      for (int i = 0; i < 4; ++i)
        #pragma unroll
        for (int j = 0; j < 2; ++j)
          acc[i][j] = __builtin_amdgcn_wmma_f32_16x16x4_f32(
              false, a[i], false, b[j], (short)0, acc[i][j], false, false);
    }
  }

  // epilogue: C layout — VGPR r holds row (r + 8*lanehalf), col = lane&15
  const int rh = (lane >> 4) * 8;
  #pragma unroll
  for (int i = 0; i < 4; ++i) {
    #pragma unroll
    for (int j = 0; j < 2; ++j) {
      int n = bn + waveN * 32 + j * 16 + idx;
      if (n >= N) continue;
      float bv = bias[n];
      #pragma unroll
      for (int r = 0; r < 8; ++r) {
        int m = bm + waveM * 64 + i * 16 + r + rh;
        if (m < M) C[(size_t)m * N + n] = acc[i][j][r] + bv;
      }
    }
  }
}

// ---------------------------------------------------------------------------
// Zero pad rows [S, SPAD) of K and V buffers.
// ---------------------------------------------------------------------------
__global__ void zero_pad_rows(float* __restrict__ Kb, float* __restrict__ Vb) {
  int n = (SPAD - S_) * DIM_;
  for (int i = blockIdx.x * blockDim.x + threadIdx.x; i < n;
       i += gridDim.x * blockDim.x) {
    Kb[(size_t)S_ * DIM_ + i] = 0.f;
    Vb[(size_t)S_ * DIM_ + i] = 0.f;
  }
}

// ---------------------------------------------------------------------------
// RMSNorm over 1536 + 3-axis RoPE (splits 22/21/21 per 64-pair head).
// One block (256 thr) per sequence position; handles Q and K rows.
// ---------------------------------------------------------------------------
__global__ __launch_bounds__(256)
void rmsnorm_rope(float* __restrict__ Q, float* __restrict__ K,
                  const float* __restrict__ nqw, const float* __restrict__ nkw,
                  const float* __restrict__ fcos, const float* __restrict__ fsin) {
  const int s   = blockIdx.x;
  const int tid = threadIdx.x;
  float* qrow = Q + (size_t)s * DIM_;
  float* krow = K + (size_t)s * DIM_;

  float sq = 0.f, sk = 0.f;
  for (int j = tid; j < DIM_; j += 256) {
    float a = qrow[j]; sq += a * a;
    float b = krow[j]; sk += b * b;
  }
  #pragma unroll
  for (int off = 16; off > 0; off >>= 1) {
    sq += __shfl_down(sq, off, 32);
    sk += __shfl_down(sk, off, 32);
  }
  __shared__ float redq[8], redk[8];
  __shared__ float srq, srk;
  if ((tid & 31) == 0) { redq[tid >> 5] = sq; redk[tid >> 5] = sk; }
  __syncthreads();
  if (tid == 0) {
    float a = 0.f, b = 0.f;
    for (int w = 0; w < 8; ++w) { a += redq[w]; b += redk[w]; }
    srq = rsqrtf(a * (1.0f / DIM_) + 1e-6f);
    srk = rsqrtf(b * (1.0f / DIM_) + 1e-6f);
  }
  __syncthreads();
  const float rq = srq, rk = srk;

  const int f   = s / 900;
  const int rem = s - f * 900;
  const int h   = rem / 30;
  const int w   = rem - h * 30;

  for (int p = tid; p < NH_ * 64; p += 256) {
    int head = p >> 6;
    int c    = p & 63;
    int row  = (c < 22) ? f : (c < 43 ? h : w);
    float cv = fcos[row * 64 + c];
    float sv = fsin[row * 64 + c];
    int b0 = head * HD_ + 2 * c;
    float xr = qrow[b0] * rq * nqw[b0];
    float xi = qrow[b0 + 1] * rq * nqw[b0 + 1];
    qrow[b0]     = xr * cv - xi * sv;
    qrow[b0 + 1] = xr * sv + xi * cv;
    xr = krow[b0] * rk * nkw[b0];
    xi = krow[b0 + 1] * rk * nkw[b0 + 1];
    krow[b0]     = xr * cv - xi * sv;
    krow[b0 + 1] = xr * sv + xi * cv;
  }
}

// ---------------------------------------------------------------------------
// Flash attention (fp32 WMMA for QK^T and PV). 4 waves/block, 16 q-rows/wave.
// K/V 16x128 tiles double-buffered in LDS via async-to-LDS copies.
// ---------------------------------------------------------------------------
#define KP 132  // pitch for 16x128 K/V tiles (16B-aligned rows, conflict-free)
#define PP 18   // pitch for 16x16 P tiles

__global__ __launch_bounds__(128)
void flash_attn(const float* __restrict__ Q, const float* __restrict__ K,
                const float* __restrict__ V, float* __restrict__ O) {
  const int head = blockIdx.y;
  const int tid  = threadIdx.x;
  const int lane = tid & 31;
  const int wid  = tid >> 5;              // 0..3
  const int q0   = blockIdx.x * 64 + wid * 16;
  const int idx  = lane & 15;
  const int half = (lane >> 4) << 1;      // 0 or 2

  __shared__ float Ks[2][16 * KP];
  __shared__ float Vs[2][16 * KP];
  __shared__ float Ps[4][16 * PP];

  // Q fragments (16 rows x 128) in registers: 32 K-steps of 4
  v2f aq[32];
  {
    int qrow = q0 + idx;
    if (qrow < S_) {
      const float* qp = Q + (size_t)qrow * DIM_ + head * HD_;
      #pragma unroll
      for (int t = 0; t < 32; ++t) {
        aq[t].x = qp[t * 4 + half];
        aq[t].y = qp[t * 4 + half + 1];
      }
    } else {
      #pragma unroll
      for (int t = 0; t < 32; ++t) { aq[t].x = 0.f; aq[t].y = 0.f; }
    }
  }

  v8f vzero = {};
  v8f outacc[8];
  #pragma unroll
  for (int j = 0; j < 8; ++j) outacc[j] = vzero;
  float mrow[8], lrow[8];
  #pragma unroll
  for (int r = 0; r < 8; ++r) { mrow[r] = -3.0e38f; lrow[r] = 0.f; }
  const float scale = 0.08838834764831845f;  // 1/sqrt(128)

  const int srow = tid >> 3;           // 0..15
  const int scb  = (tid & 7) * 16;     // 0..112
  auto issue_kv = [&](int kt, int buf) {
    const float* kp = K + (size_t)(kt + srow) * DIM_ + head * HD_ + scb;
    const float* vp = V + (size_t)(kt + srow) * DIM_ + head * HD_ + scb;
    #pragma unroll
    for (int u = 0; u < 4; ++u) {
      async_copy16(kp + u * 4, &Ks[buf][srow * KP + scb + u * 4]);
      async_copy16(vp + u * 4, &Vs[buf][srow * KP + scb + u * 4]);
    }
  };

  issue_kv(0, 0);

  for (int t = 0; t < SPAD / 16; ++t) {
    const int kt = t * 16;
    WAIT_ASYNC();        // own tile-t copies done
    __syncthreads();     // all copies visible; prior PV on other buf drained
    if (kt + 16 < SPAD) issue_kv(kt + 16, (t + 1) & 1);  // prefetch next tile

    const float* Kb = Ks[t & 1];
    const float* Vb = Vs[t & 1];

    // S-tile = Q K^T (16x16), K reduced over 128 in 32 WMMA steps
    v8f stile = vzero;
    #pragma unroll
    for (int u = 0; u < 32; ++u) {
      v2f b;
      b.x = Kb[idx * KP + u * 4 + half];
      b.y = Kb[idx * KP + u * 4 + half + 1];
      stile = __builtin_amdgcn_wmma_f32_16x16x4_f32(
          false, aq[u], false, b, (short)0, stile, false, false);
    }

    // scale + mask invalid key columns
    const bool valid = (kt + idx) < S_;
    float pv[8];
    #pragma unroll
    for (int r = 0; r < 8; ++r) {
      float v = stile[r] * scale;
      pv[r] = valid ? v : -3.0e38f;
    }

    // online softmax: row reductions across each 16-lane half
    #pragma unroll
    for (int r = 0; r < 8; ++r) {
      float mx = pv[r];
      mx = fmaxf(mx, __shfl_xor(mx, 1, 32));
      mx = fmaxf(mx, __shfl_xor(mx, 2, 32));
      mx = fmaxf(mx, __shfl_xor(mx, 4, 32));
      mx = fmaxf(mx, __shfl_xor(mx, 8, 32));
      float mnew  = fmaxf(mrow[r], mx);
      float alpha = __expf(mrow[r] - mnew);
      float p     = __expf(pv[r] - mnew);
      float rs = p;
      rs += __shfl_xor(rs, 1, 32);
      rs += __shfl_xor(rs, 2, 32);
      rs += __shfl_xor(rs, 4, 32);
      rs += __shfl_xor(rs, 8, 32);
      lrow[r] = lrow[r] * alpha + rs;
      mrow[r] = mnew;
      pv[r]   = p;
      #pragma unroll
      for (int j = 0; j < 8; ++j) outacc[j][r] *= alpha;
    }

    // C-layout -> A-layout for P via wave-private LDS slab
    #pragma unroll
    for (int r = 0; r < 8; ++r)
      Ps[wid][(r + (half << 2)) * PP + idx] = pv[r];
    __syncthreads();

    // PV: out(16x128) += P(16x16) @ V(16x128)
    #pragma unroll
    for (int kk = 0; kk < 16; kk += 4) {
      v2f a;
      a.x = Ps[wid][idx * PP + kk + half];
      a.y = Ps[wid][idx * PP + kk + half + 1];
      #pragma unroll
      for (int j = 0; j < 8; ++j) {
        v2f b;
        b.x = Vb[(kk + half) * KP + j * 16 + idx];
        b.y = Vb[(kk + half + 1) * KP + j * 16 + idx];
        outacc[j] = __builtin_amdgcn_wmma_f32_16x16x4_f32(
            false, a, false, b, (short)0, outacc[j], false, false);
      }
    }
  }

  // finalize: divide by softmax denominator, store (S, NH*HD)
  const int rh = (lane >> 4) * 8;
  #pragma unroll
  for (int j = 0; j < 8; ++j) {
    int n = head * HD_ + j * 16 + idx;
    #pragma unroll
    for (int r = 0; r < 8; ++r) {
      int m = q0 + r + rh;
      if (m < S_) O[(size_t)m * DIM_ + n] = outacc[j][r] / lrow[r];
    }
  }
}

// ---------------------------------------------------------------------------
extern "C" void kernel_launch(void* const* d_in, const int* in_sizes, int n_in,
                              void* d_out, int out_size, void* d_ws, size_t ws_size,
                              hipStream_t stream) {
  (void)in_sizes; (void)n_in; (void)out_size; (void)ws_size;
  const float* x    = (const float*)d_in[0];
  const float* q_w  = (const float*)d_in[1];
  const float* q_b  = (const float*)d_in[2];
  const float* k_w  = (const float*)d_in[3];
  const float* k_b  = (const float*)d_in[4];
  const float* v_w  = (const float*)d_in[5];
  const float* v_b  = (const float*)d_in[6];
  const float* o_w  = (const float*)d_in[7];
  const float* o_b  = (const float*)d_in[8];
  const float* nqw  = (const float*)d_in[9];
  const float* nkw  = (const float*)d_in[10];
  const float* fcos = (const float*)d_in[11];
  const float* fsin = (const float*)d_in[12];
  float* out = (float*)d_out;

  float* Qb = (float*)d_ws;                    // S    x 1536
  float* Kb = Qb + (size_t)S_   * DIM_;        // SPAD x 1536 (zero-padded)
  float* Vb = Kb + (size_t)SPAD * DIM_;        // SPAD x 1536 (zero-padded)
  float* Ab = Vb + (size_t)SPAD * DIM_;        // S    x 1536

  dim3 gblk(256);
  dim3 ggrid((S_ + 127) / 128, DIM_ / 128);
  gemm_xwT_bias<<<ggrid, gblk, 0, stream>>>(x, q_w, q_b, Qb, S_, DIM_, DIM_);
  gemm_xwT_bias<<<ggrid, gblk, 0, stream>>>(x, k_w, k_b, Kb, S_, DIM_, DIM_);
  gemm_xwT_bias<<<ggrid, gblk, 0, stream>>>(x, v_w, v_b, Vb, S_, DIM_, DIM_);
  zero_pad_rows<<<64, 256, 0, stream>>>(Kb, Vb);
  rmsnorm_rope<<<S_, 256, 0, stream>>>(Qb, Kb, nqw, nkw, fcos, fsin);
  flash_attn<<<dim3((S_ + 63) / 64, NH_), 128, 0, stream>>>(Qb, Kb, Vb, Ab);
  gemm_xwT_bias<<<ggrid, gblk, 0, stream>>>(Ab, o_w, o_b, out, S_, DIM_, DIM_);
}